// Get_PPR_24154896073102
// MI455X (gfx1250) — compile-verified
//
#include <hip/hip_runtime.h>
#include <hip/hip_bf16.h>
#include <stdint.h>
#include <stddef.h>

// ---------------- constants from the reference ----------------
#define PPR_ALPHA   0.15f
#define PPR_RA      (1e-4f * 0.15f)           // RHO*ALPHA
#define PPR_THRESH  ((1.0f + 0.01f) * PPR_RA) // (1+eps)*rho*alpha
#define PPR_HALF_C  (0.5f * (1.0f - 0.15f))   // 0.5*(1-alpha)
#define PPR_NITER   20
#define PPR_NS      8                         // seeds fused in innermost dim

#if defined(__gfx1250__) && __has_builtin(__builtin_amdgcn_global_load_async_to_lds_b128)
#define USE_ASYNC_128 1
#else
#define USE_ASYNC_128 0
#endif

typedef int v4i_t __attribute__((vector_size(4 * sizeof(int))));
typedef __attribute__((address_space(1))) v4i_t g1_v4i;
typedef __attribute__((address_space(3))) v4i_t l3_v4i;

__device__ __forceinline__ void wait_async0() {
#if defined(__gfx1250__)
#if __has_builtin(__builtin_amdgcn_s_wait_asynccnt)
  __builtin_amdgcn_s_wait_asynccnt(0);
#else
  asm volatile("s_wait_asynccnt 0" ::: "memory");
#endif
#endif
}

// ---------------- init: p=0, d=seed impulse, tmp=0 ----------------
__global__ void ppr_init(float* __restrict__ p, float* __restrict__ d,
                         float* __restrict__ tmp, unsigned* __restrict__ flags,
                         const float* __restrict__ deg,
                         const int* __restrict__ seeds, int n) {
  int idx = blockIdx.x * blockDim.x + threadIdx.x;
  int total = n * PPR_NS;
  if (idx < 8) flags[idx] = 0u;
  if (idx >= total) return;
  int node = idx >> 3, s = idx & 7;
  p[idx] = 0.0f;
  tmp[idx] = 0.0f;
  float dv = 0.0f;
  if (node == seeds[s]) {
    float di = 1.0f / fmaxf(deg[node], 1e-12f);
    dv = -PPR_ALPHA * di;
  }
  d[idx] = dv;
}

// ---------------- per-iteration flag reset ----------------
__global__ void ppr_reset_flags(unsigned* __restrict__ flags) {
  if (threadIdx.x < 8) flags[threadIdx.x] = 0u;
}

// ---------------- phase A: frontier + interleaved gather table + max|d| ----------------
// pg layout per node: [dpk_s0..dpk_s7, g_s0..g_s7]  (64 contiguous bytes)
__global__ void ppr_phaseA(const float* __restrict__ p, const float* __restrict__ d,
                           const float* __restrict__ deg, float* __restrict__ pg,
                           unsigned* __restrict__ flags, int n) {
  int idx = blockIdx.x * blockDim.x + threadIdx.x;
  int total = n * PPR_NS;
  float absd = 0.0f;
  if (idx < total) {
    int node = idx >> 3, s = idx & 7;
    float pv = p[idx], dv = d[idx];
    bool S = (pv - dv) >= PPR_RA;
    float dpk = S ? -(dv + PPR_RA) : 0.0f;
    float g   = S ? deg[node] : 0.0f;
    size_t b = (size_t)node * 16 + (size_t)s;
    pg[b]     = dpk;
    pg[b + 8] = g;
    absd = fabsf(dv);
  }
  // per-seed block max reduction (seed = tid&7 invariant for strides >= 8)
  __shared__ float red[256];
  red[threadIdx.x] = absd;
  __syncthreads();
  for (int off = 128; off >= 8; off >>= 1) {
    if (threadIdx.x < (unsigned)off)
      red[threadIdx.x] = fmaxf(red[threadIdx.x], red[threadIdx.x + off]);
    __syncthreads();
  }
  if (threadIdx.x < 8)
    atomicMax(&flags[threadIdx.x], __float_as_uint(red[threadIdx.x]));
}

// ---------------- phase B: 4 edges/thread push ----------------
// gather 64B/edge from pg (L2-resident), scatter via f32 atomics into tmp
__global__ void ppr_edges(const int* __restrict__ row, const int* __restrict__ col,
                          const float* __restrict__ val,
                          const float* __restrict__ pg,
                          const unsigned* __restrict__ flags,
                          float* __restrict__ tmp, int nedges) {
  // per-seed active flags (uniform -> scalar loads); whole-grid early exit once converged
  bool act[8];
  bool any = false;
#pragma unroll
  for (int s = 0; s < 8; ++s) {
    act[s] = __uint_as_float(flags[s]) > PPR_THRESH;
    any = any || act[s];
  }
  if (!any) return;

  int tid = threadIdx.x;
  int base = (blockIdx.x * blockDim.x + tid) * 4;
  if (base >= nedges) return;
  int nval = nedges - base;
  if (nval > 4) nval = 4;
  bool full = (nval == 4);

  int r[4], c[4];
  float vv[4];

#if USE_ASYNC_128
  // gfx1250 async copy: stage 4 row + 4 col indices per lane through LDS (ASYNCcnt)
  __shared__ int lrow[1024];
  __shared__ int lcol[1024];
  if (full) {
    __builtin_amdgcn_global_load_async_to_lds_b128(
        (g1_v4i*)(row + base), (l3_v4i*)&lrow[tid * 4], 0, 0);
    __builtin_amdgcn_global_load_async_to_lds_b128(
        (g1_v4i*)(col + base), (l3_v4i*)&lcol[tid * 4], 0, 0);
  }
  wait_async0();  // lane reads only its own slots -> no block barrier needed
  if (full) {
#pragma unroll
    for (int j = 0; j < 4; ++j) {
      r[j] = lrow[tid * 4 + j];
      c[j] = lcol[tid * 4 + j];
    }
  }
#else
  if (full) {
    int4 r4 = *(const int4*)(row + base);
    int4 c4 = *(const int4*)(col + base);
    r[0] = r4.x; r[1] = r4.y; r[2] = r4.z; r[3] = r4.w;
    c[0] = c4.x; c[1] = c4.y; c[2] = c4.z; c[3] = c4.w;
  }
#endif
  if (full) {
    float4 v4 = *(const float4*)(val + base);
    vv[0] = v4.x; vv[1] = v4.y; vv[2] = v4.z; vv[3] = v4.w;
  } else {
    for (int j = 0; j < nval; ++j) {
      r[j] = row[base + j];
      c[j] = col[base + j];
      vv[j] = val[base + j];
    }
  }

  // stream-ahead prefetch of the edge arrays (global_prefetch_b8)
  const int PF = 8192;
  if (base + PF < nedges) {
    __builtin_prefetch(row + base + PF, 0, 0);
    __builtin_prefetch(col + base + PF, 0, 0);
    __builtin_prefetch(val + base + PF, 0, 0);
  }

  for (int j = 0; j < nval; ++j) {
    const float4* P = (const float4*)(pg + (size_t)c[j] * 16);
    float4 a0 = P[0], a1 = P[1];   // dpk for seeds 0..7
    float4 g0 = P[2], g1 = P[3];   // gated degree for seeds 0..7
    float dpk[8] = {a0.x, a0.y, a0.z, a0.w, a1.x, a1.y, a1.z, a1.w};
    float gg[8]  = {g0.x, g0.y, g0.z, g0.w, g1.x, g1.y, g1.z, g1.w};
    float v = vv[j];
    float* t = tmp + (size_t)r[j] * PPR_NS;
#pragma unroll
    for (int s = 0; s < 8; ++s) {
      if (!act[s]) continue;
      float num = v * dpk[s];
      float den = fmaf(v, gg[s], 1e-12f);
      float q = __fdividef(num, den);
      if (q != 0.0f) atomicAdd(&t[s], q);  // frontier is sparse: most atomics skipped
    }
  }
}

// ---------------- phase C: gated node update; also re-zero tmp ----------------
__global__ void ppr_phaseC(float* __restrict__ p, float* __restrict__ d,
                           const float* __restrict__ deg, float* __restrict__ tmp,
                           const unsigned* __restrict__ flags, int n) {
  int idx = blockIdx.x * blockDim.x + threadIdx.x;
  int total = n * PPR_NS;
  if (idx >= total) return;
  int node = idx >> 3, s = idx & 7;
  float tv = tmp[idx];
  tmp[idx] = 0.0f;                                        // ready for next iteration
  if (!(__uint_as_float(flags[s]) > PPR_THRESH)) return;  // inactive seed: p,d frozen
  float pv = p[idx], dv = d[idx];
  float di = 1.0f / fmaxf(deg[node], 1e-12f);
  float half = PPR_HALF_C * di;
  bool S = (pv - dv) >= PPR_RA;
  float dpk = S ? -(dv + PPR_RA) : 0.0f;
  float dn, pn;
  if (S) {
    dn = (1.0f - di) * dv - PPR_RA * di - half * dpk - half * tv;
    pn = pv + dpk;
  } else {
    dn = dv - half * tv;
    pn = pv;
  }
  d[idx] = dn;
  p[idx] = pn;
}

// ---------------- output transpose: [node][8] -> [8][n] ----------------
__global__ void ppr_emit(const float* __restrict__ p, float* __restrict__ out, int n) {
  int idx = blockIdx.x * blockDim.x + threadIdx.x;
  int total = n * PPR_NS;
  if (idx >= total) return;
  int s = idx / n, node = idx - s * n;
  out[idx] = p[(size_t)node * PPR_NS + s];
}

// ---------------- launch ----------------
extern "C" void kernel_launch(void* const* d_in, const int* in_sizes, int n_in,
                              void* d_out, int out_size, void* d_ws, size_t ws_size,
                              hipStream_t stream) {
  const int*   row   = (const int*)d_in[0];
  const int*   col   = (const int*)d_in[1];
  const float* val   = (const float*)d_in[2];
  const float* deg   = (const float*)d_in[3];
  const int*   seeds = (const int*)d_in[4];
  const int E = in_sizes[0];
  const int n = in_sizes[3];
  float* out = (float*)d_out;

  // workspace layout (floats): p[8n] | d[8n] | pg[16n] | tmp[8n] | flags[8 u32]
  float* ws  = (float*)d_ws;
  float* p   = ws;
  float* d   = ws + (size_t)8 * n;
  float* pg  = ws + (size_t)16 * n;
  float* tmp = ws + (size_t)32 * n;
  unsigned* flags = (unsigned*)(ws + (size_t)40 * n);

  const int threads = 256;
  const int blocksN = (n * PPR_NS + threads - 1) / threads;
  const int edgeThreads = (E + 3) / 4;
  const int blocksE = (edgeThreads + threads - 1) / threads;

  ppr_init<<<blocksN, threads, 0, stream>>>(p, d, tmp, flags, deg, seeds, n);

  for (int it = 0; it < PPR_NITER; ++it) {
    ppr_reset_flags<<<1, 32, 0, stream>>>(flags);
    ppr_phaseA<<<blocksN, threads, 0, stream>>>(p, d, deg, pg, flags, n);
    ppr_edges<<<blocksE, threads, 0, stream>>>(row, col, val, pg, flags, tmp, E);
    ppr_phaseC<<<blocksN, threads, 0, stream>>>(p, d, deg, tmp, flags, n);
  }

  ppr_emit<<<blocksN, threads, 0, stream>>>(p, out, n);
}